// GCN_22883585753783
// MI455X (gfx1250) — compile-verified
//
#include <hip/hip_runtime.h>
#include <hip/hip_bf16.h>
#include <stdint.h>

typedef __attribute__((ext_vector_type(2))) float v2f;
typedef __attribute__((ext_vector_type(8))) float v8f;

#define CH 64           // IN_CH == HID_CH == OUT_CH == 64
#define NGRAPH 128

// ---------------------------------------------------------------------------
// Degree / norm kernels
// ---------------------------------------------------------------------------
__global__ void k_fill_deg(float* __restrict__ deg, int n) {
    int i = blockIdx.x * blockDim.x + threadIdx.x;
    if (i < n) deg[i] = 1.0f;            // self-loop contributes 1
}

__global__ void k_deg_edges(float* __restrict__ deg,
                            const long long* __restrict__ dst, int e) {
    int i = blockIdx.x * blockDim.x + threadIdx.x;
    if (i < e) atomicAdd(&deg[(int)dst[i]], 1.0f);
}

__global__ void k_dinv(float* __restrict__ deg, int n) {
    int i = blockIdx.x * blockDim.x + threadIdx.x;
    if (i < n) deg[i] = rsqrtf(deg[i]);
}

// ---------------------------------------------------------------------------
// WMMA GEMM: out[M,64] = op(A)[M,64] @ B[64,64] (+bias, +relu_out)
// One wave computes a 16x16 tile via 16 chained V_WMMA_F32_16X16X4_F32.
// 256-thread block = 8 waves = 32(M) x 64(N) per block. M must be %32 == 0.
// No divergence around WMMA -> EXEC all ones (ISA requirement).
//
// FUSE_INIT: epilogue additionally writes the scatter accumulator
//   acc[row][c] = biasAcc[c] + out[row][c] * dinv[row]^2
// A-matrix rows are read only by the block that owns them, so a block-wide
// barrier between the K-loop and the epilogue makes acc==A aliasing safe.
// ---------------------------------------------------------------------------
template <bool RELU_IN, bool BIAS, bool RELU_OUT, bool FUSE_INIT>
__global__ void k_gemm_wmma(const float* __restrict__ A,
                            const float* __restrict__ B,
                            const float* __restrict__ bias,
                            float* __restrict__ out,
                            float* __restrict__ acc,        // FUSE_INIT only
                            const float* __restrict__ dinv, // FUSE_INIT only
                            const float* __restrict__ biasAcc, // FUSE_INIT only
                            int M) {
    const int wave = threadIdx.x >> 5;          // 0..7
    const int lane = threadIdx.x & 31;
    const int mtile = blockIdx.x * 2 + (wave >> 2);
    const int ntile = wave & 3;
    const int m0 = mtile * 16;
    const int n0 = ntile * 16;
    const int half = lane >> 4;                 // 0/1: selects K pair
    const int l15  = lane & 15;

    const float* Arow = A + (size_t)(m0 + l15) * CH; // lane's A row
    v8f c = {0.f, 0.f, 0.f, 0.f, 0.f, 0.f, 0.f, 0.f};

#pragma unroll
    for (int k = 0; k < CH; k += 4) {
        v2f a = *(const v2f*)(Arow + k + 2 * half);
        if (RELU_IN) { a.x = fmaxf(a.x, 0.f); a.y = fmaxf(a.y, 0.f); }
        v2f b;
        b.x = B[(size_t)(k + 2 * half + 0) * CH + n0 + l15];
        b.y = B[(size_t)(k + 2 * half + 1) * CH + n0 + l15];
        // D = A(16x4) x B(4x16) + C   -> v_wmma_f32_16x16x4_f32
        c = __builtin_amdgcn_wmma_f32_16x16x4_f32(
                /*neg_a=*/false, a, /*neg_b=*/false, b,
                /*c_mod=*/(short)0, c, /*reuse_a=*/false, /*reuse_b=*/false);
    }

    if (FUSE_INIT) __syncthreads();  // all A-row reads in block done before acc writes

    const int coln = n0 + l15;
    float bv  = BIAS ? bias[coln] : 0.f;
    float bav = FUSE_INIT ? biasAcc[coln] : 0.f;
#pragma unroll
    for (int r = 0; r < 8; ++r) {
        int row = m0 + r + 8 * half;
        float v = c[r] + bv;
        if (RELU_OUT) v = fmaxf(v, 0.f);
        out[(size_t)row * CH + coln] = v;
        if (FUSE_INIT) {
            float di = dinv[row];
            acc[(size_t)row * CH + coln] = bav + v * (di * di);
        }
    }
    (void)M;
}

// ---------------------------------------------------------------------------
// Edge scatter: one wave per edge, 2 channels per lane.
// Gather of xw[src] row is one coalesced 256B access; atomics hit L2.
// ---------------------------------------------------------------------------
__global__ void k_scatter(float* __restrict__ out,
                          const float* __restrict__ xw,
                          const float* __restrict__ dinv,
                          const long long* __restrict__ src,
                          const long long* __restrict__ dst, int e) {
    int gid = blockIdx.x * blockDim.x + threadIdx.x;
    int edge = gid >> 5;
    if (edge >= e) return;
    int cp = (gid & 31) * 2;
    int s = (int)src[edge];
    int d = (int)dst[edge];
    float norm = dinv[s] * dinv[d];
    v2f v = *(const v2f*)(xw + (size_t)s * CH + cp);   // global_load_b64
    float* orow = out + (size_t)d * CH + cp;
    atomicAdd(orow + 0, v.x * norm);
    atomicAdd(orow + 1, v.y * norm);
}

// ---------------------------------------------------------------------------
// Pooling: batch ids are sorted -> keep running partial per thread, flush
// an atomic only on graph-id change. ReLU fused on load.
// Block covers 64 nodes; thread t: channel = t&63, node stride 4.
// ---------------------------------------------------------------------------
__global__ void k_zero_pool(float* __restrict__ pooled, float* __restrict__ counts) {
    int i = blockIdx.x * blockDim.x + threadIdx.x;
    if (i < NGRAPH * CH) pooled[i] = 0.f;
    if (i < NGRAPH) counts[i] = 0.f;
}

__global__ void k_counts(const long long* __restrict__ batch,
                         float* __restrict__ counts, int n) {
    int i = blockIdx.x * blockDim.x + threadIdx.x;
    if (i < n) atomicAdd(&counts[(int)batch[i]], 1.0f);
}

__global__ void k_pool(const float* __restrict__ h,
                       const long long* __restrict__ batch,
                       float* __restrict__ pooled, int n) {
    int c = threadIdx.x & 63;
    int sub = threadIdx.x >> 6;                 // 0..3
    int nbase = blockIdx.x * 64;
    float acc = 0.f;
    int curg = -1;
    for (int j = sub; j < 64; j += 4) {
        int node = nbase + j;
        if (node >= n) break;
        int g = (int)batch[node];
        if (g != curg) {
            if (curg >= 0) atomicAdd(&pooled[(size_t)curg * CH + c], acc);
            curg = g; acc = 0.f;
        }
        acc += fmaxf(h[(size_t)node * CH + c], 0.f);
    }
    if (curg >= 0) atomicAdd(&pooled[(size_t)curg * CH + c], acc);
}

__global__ void k_normalize(float* __restrict__ pooled,
                            const float* __restrict__ counts) {
    int i = blockIdx.x * blockDim.x + threadIdx.x;
    if (i < NGRAPH * CH) pooled[i] /= fmaxf(counts[i >> 6], 1.0f);
}

// ---------------------------------------------------------------------------
// Launch
// ---------------------------------------------------------------------------
extern "C" void kernel_launch(void* const* d_in, const int* in_sizes, int n_in,
                              void* d_out, int out_size, void* d_ws, size_t ws_size,
                              hipStream_t stream) {
    const float*     x    = (const float*)d_in[0];
    const long long* ei   = (const long long*)d_in[1];   // [2, E] int64
    const long long* bat  = (const long long*)d_in[2];   // [N] int64
    const float*     W1   = (const float*)d_in[3];
    const float*     b1   = (const float*)d_in[4];
    const float*     W2   = (const float*)d_in[5];
    const float*     b2   = (const float*)d_in[6];
    const float*     Wfc  = (const float*)d_in[7];
    const float*     bfc  = (const float*)d_in[8];
    float*           outp = (float*)d_out;

    const int N = in_sizes[0] / CH;      // 100000
    const int E = in_sizes[1] / 2;       // 1250000
    const long long* src = ei;
    const long long* dst = ei + E;

    // workspace carve-up (~52 MB)
    float* buf0   = (float*)d_ws;                 // N*64  (xw)
    float* buf1   = buf0 + (size_t)N * CH;        // N*64  (accumulator / h)
    float* dinv   = buf1 + (size_t)N * CH;        // N
    float* pooled = dinv + N;                     // 128*64
    float* counts = pooled + NGRAPH * CH;         // 128

    const int T = 256;
    dim3 blk(T);

    // degrees -> dinv
    k_fill_deg<<<dim3((N + T - 1) / T), blk, 0, stream>>>(dinv, N);
    k_deg_edges<<<dim3((E + T - 1) / T), blk, 0, stream>>>(dinv, dst, E);
    k_dinv<<<dim3((N + T - 1) / T), blk, 0, stream>>>(dinv, N);

    const int gemmBlocksN = N / 32;      // 100000/32 = 3125 exact

    // layer 1: xw = x @ W1 ; fused acc init: buf1 = b1 + xw*dinv^2
    k_gemm_wmma<false, false, false, true><<<dim3(gemmBlocksN), blk, 0, stream>>>(
        x, W1, nullptr, buf0, buf1, dinv, b1, N);
    // scatter real edges into buf1
    k_scatter<<<dim3(((size_t)E * 32 + T - 1) / T), blk, 0, stream>>>(
        buf1, buf0, dinv, src, dst, E);

    // layer 2: xw2 = relu(h1) @ W2 ; fused acc init overwrites buf1 (barrier-safe)
    k_gemm_wmma<true, false, false, true><<<dim3(gemmBlocksN), blk, 0, stream>>>(
        buf1, W2, nullptr, buf0, buf1, dinv, b2, N);
    k_scatter<<<dim3(((size_t)E * 32 + T - 1) / T), blk, 0, stream>>>(
        buf1, buf0, dinv, src, dst, E);

    // pooling (ReLU fused on load), counts, normalize
    k_zero_pool<<<dim3((NGRAPH * CH + T - 1) / T), blk, 0, stream>>>(pooled, counts);
    k_counts<<<dim3((N + T - 1) / T), blk, 0, stream>>>(bat, counts, N);
    k_pool<<<dim3((N + 63) / 64), blk, 0, stream>>>(buf1, bat, pooled, N);
    k_normalize<<<dim3((NGRAPH * CH + T - 1) / T), blk, 0, stream>>>(pooled, counts);

    // final FC: out = relu(pooled @ Wfc + bfc), M = 128 -> 4 blocks
    k_gemm_wmma<false, true, true, false><<<dim3(NGRAPH / 32), blk, 0, stream>>>(
        pooled, Wfc, bfc, outp, nullptr, nullptr, nullptr, NGRAPH);
    (void)n_in; (void)out_size; (void)ws_size;
}